// SerialSum_61933428409683
// MI455X (gfx1250) — compile-verified
//
#include <hip/hip_runtime.h>
#include <hip/hip_bf16.h>

typedef float v2f __attribute__((ext_vector_type(2)));
typedef float v4f __attribute__((ext_vector_type(4)));
typedef float v8f __attribute__((ext_vector_type(8)));

static __device__ __forceinline__ v8f wmma_acc(float lo, float hi, v8f c) {
  v2f a;
  a.x = lo;
  a.y = hi;
  v2f ones;
  ones.x = 1.0f;
  ones.y = 1.0f;
  // 8 args: (neg_a, A, neg_b, B, c_mod, C, reuse_a, reuse_b)
  return __builtin_amdgcn_wmma_f32_16x16x4_f32(false, a, false, ones,
                                               (short)0, c, false, false);
}

// Stage 1: stream x with NT float4 loads, accumulate via V_WMMA_F32_16X16X4_F32
// with B = ones (D[r][c] += sum_k A[r][k]); sum(D) == 16 * true_sum, /16 is
// exact. Trip count is scalar-uniform so EXEC stays all-1s around every WMMA.
__global__ void __launch_bounds__(256) serialsum_stage1(
    const float* __restrict__ x, long long n, long long n4,
    float* __restrict__ partials) {
  v8f d0 = {}, d1 = {}, d2 = {}, d3 = {};

  const v4f* __restrict__ p = (const v4f*)x;
  const long long tid    = (long long)blockIdx.x * blockDim.x + threadIdx.x;
  const long long stride = (long long)gridDim.x * blockDim.x;
  const long long full   = n4 / stride;  // uniform for every thread in the grid

  long long i  = tid;
  long long it = 0;

  // 4x unrolled: 4 B128 NT loads in flight per wave, 8 WMMAs over 4 accums.
  for (; it + 4 <= full; it += 4) {
    v4f va = __builtin_nontemporal_load(&p[i]);
    v4f vb = __builtin_nontemporal_load(&p[i + stride]);
    v4f vc = __builtin_nontemporal_load(&p[i + 2 * stride]);
    v4f vd = __builtin_nontemporal_load(&p[i + 3 * stride]);
    i += 4 * stride;
    d0 = wmma_acc(va.x, va.y, d0);
    d1 = wmma_acc(va.z, va.w, d1);
    d2 = wmma_acc(vb.x, vb.y, d2);
    d3 = wmma_acc(vb.z, vb.w, d3);
    d0 = wmma_acc(vc.x, vc.y, d0);
    d1 = wmma_acc(vc.z, vc.w, d1);
    d2 = wmma_acc(vd.x, vd.y, d2);
    d3 = wmma_acc(vd.z, vd.w, d3);
  }
  for (; it < full; ++it) {
    v4f v = __builtin_nontemporal_load(&p[i]);
    i += stride;
    d0 = wmma_acc(v.x, v.y, d0);
    d1 = wmma_acc(v.z, v.w, d1);
  }

  // Lanes 0-15 hold D rows M=0..7 (one per VGPR), lanes 16-31 hold M=8..15,
  // column N = lane%16: summing 8 VGPRs per lane then across 32 lanes covers
  // every D entry exactly once -> 16 * partial_sum.
  v8f d = (d0 + d1) + (d2 + d3);
  float s = 0.0f;
#pragma unroll
  for (int k = 0; k < 8; ++k) s += d[k];
  s *= 0.0625f;  // exact /16

  // Residual float4s (n4 % stride) — plain VALU, divergence legal here.
  float extra = 0.0f;
  if (i < n4) {
    v4f v = __builtin_nontemporal_load(&p[i]);
    extra = (v.x + v.y) + (v.z + v.w);
  }
  // Scalar tail (n % 4) — one thread.
  if (blockIdx.x == 0 && threadIdx.x == 0) {
    for (long long j = n4 * 4; j < n; ++j) extra += x[j];
  }
  s += extra;

  // wave32 reduction
#pragma unroll
  for (int off = 16; off > 0; off >>= 1) s += __shfl_down(s, off, 32);

  __shared__ float lds[8];
  const int lane = threadIdx.x & 31;
  const int wid  = threadIdx.x >> 5;
  if (lane == 0) lds[wid] = s;
  __syncthreads();
  if (threadIdx.x == 0) {
    float t = 0.0f;
#pragma unroll
    for (int w = 0; w < 8; ++w) t += lds[w];
    partials[blockIdx.x] = t;
  }
}

// Stage 2: single block folds the per-block partials into d_out[0].
__global__ void __launch_bounds__(256) serialsum_stage2(
    const float* __restrict__ partials, int nparts, float* __restrict__ out) {
  float s = 0.0f;
  for (int i = threadIdx.x; i < nparts; i += blockDim.x) s += partials[i];
#pragma unroll
  for (int off = 16; off > 0; off >>= 1) s += __shfl_down(s, off, 32);

  __shared__ float lds[8];
  const int lane = threadIdx.x & 31;
  const int wid  = threadIdx.x >> 5;
  if (lane == 0) lds[wid] = s;
  __syncthreads();
  if (threadIdx.x == 0) {
    float t = 0.0f;
#pragma unroll
    for (int w = 0; w < 8; ++w) t += lds[w];
    out[0] = t;
  }
}

extern "C" void kernel_launch(void* const* d_in, const int* in_sizes, int n_in,
                              void* d_out, int out_size, void* d_ws, size_t ws_size,
                              hipStream_t stream) {
  const float* x = (const float*)d_in[0];
  float* out = (float*)d_out;
  float* ws = (float*)d_ws;

  const long long n  = (long long)in_sizes[0];
  const long long n4 = n >> 2;

  int blocks = 2048;  // 2048 partials = 8 KiB of workspace
  const long long max_parts = (long long)(ws_size / sizeof(float));
  if (max_parts < blocks) blocks = (int)(max_parts > 0 ? max_parts : 1);

  serialsum_stage1<<<blocks, 256, 0, stream>>>(x, n, n4, ws);
  serialsum_stage2<<<1, 256, 0, stream>>>(ws, blocks, out);
}